// NetTGCNBasic_34668976014110
// MI455X (gfx1250) — compile-verified
//
#include <hip/hip_runtime.h>
#include <hip/hip_bf16.h>

// Problem constants (from reference)
#define NN   20000
#define BB   8
#define HH   15
#define HP   16                  // H padded to 16 for WMMA K-dim
#define KK   25
#define GG   64
#define EE   640000
#define CC   6
#define BHP  (BB*HP)             // 128 floats per node (padded)
#define MM   (NN*BB)             // 160,000 rows of the dense GEMM
#define MP   ((long)MM*HP)       // padded Tx buffer size = 2,560,000
#define NBG  ((long)NN*BB*GG)    // 10,240,000
#define NG   ((long)NN*GG)       // 1,280,000

typedef __attribute__((ext_vector_type(2))) float v2f;
typedef __attribute__((ext_vector_type(4))) float v4f;
typedef __attribute__((ext_vector_type(8))) float v8f;

// ---------------- utility kernels ----------------

__global__ void k_zero(float* __restrict__ p, long n) {
  long i = (long)blockIdx.x * blockDim.x + threadIdx.x;
  long stride = (long)gridDim.x * blockDim.x;
  for (; i < n; i += stride) p[i] = 0.0f;
}

// dst = alpha * src  (pads stay zero: alpha*0 == 0)
__global__ void k_scalecopy(float* __restrict__ dst, const float* __restrict__ src,
                            float alpha, long n) {
  long i = (long)blockIdx.x * blockDim.x + threadIdx.x;
  long stride = (long)gridDim.x * blockDim.x;
  for (; i < n; i += stride) dst[i] = alpha * src[i];
}

// deg[row[e]] += 1
__global__ void k_deg(const int* __restrict__ row, float* __restrict__ deg, int e_cnt) {
  int i = blockIdx.x * blockDim.x + threadIdx.x;
  int stride = gridDim.x * blockDim.x;
  for (; i < e_cnt; i += stride) atomicAdd(&deg[row[i]], 1.0f);
}

__global__ void k_dinv(const float* __restrict__ deg, float* __restrict__ dinv, int n) {
  int i = blockIdx.x * blockDim.x + threadIdx.x;
  if (i < n) {
    float d = deg[i];
    dinv[i] = (d > 0.0f) ? rsqrtf(fmaxf(d, 1.0f)) : 0.0f;
  }
}

__global__ void k_norm(const int* __restrict__ row, const int* __restrict__ col,
                       const float* __restrict__ dinv, float* __restrict__ norm, int e_cnt) {
  int i = blockIdx.x * blockDim.x + threadIdx.x;
  int stride = gridDim.x * blockDim.x;
  for (; i < e_cnt; i += stride) norm[i] = -(dinv[row[i]] * dinv[col[i]]);
}

// x[B,N,H=15] -> T0[N,B,HP=16]  (pad element h=15 set to zero)
__global__ void k_transpose(const float* __restrict__ x, float* __restrict__ t0) {
  long i = (long)blockIdx.x * blockDim.x + threadIdx.x;
  long stride = (long)gridDim.x * blockDim.x;
  for (; i < MP; i += stride) {
    int h = (int)(i & 15);
    long m = i >> 4;            // m = n*8 + b
    int b = (int)(m & 7);
    long n = m >> 3;
    t0[i] = (h < HH) ? x[((long)b * NN + n) * HH + h] : 0.0f;
  }
}

// W[K,15,G] -> Wp[K,16,G] with h=15 row zeroed
__global__ void k_padW(const float* __restrict__ W, float* __restrict__ Wp) {
  int i = blockIdx.x * blockDim.x + threadIdx.x;
  if (i < KK * HP * GG) {
    int k = i >> 10;            // HP*GG = 1024
    int rem = i & 1023;
    int h = rem >> 6;
    int g = rem & 63;
    Wp[i] = (h < HH) ? W[(k * HH + h) * GG + g] : 0.0f;
  }
}

// dst[row[e], :] += scale * norm[e] * src[col[e], :]   (128-float padded node blocks)
// Each work item: one edge x 4 consecutive floats, vector b128 source load.
__global__ void k_prop(const int* __restrict__ row, const int* __restrict__ col,
                       const float* __restrict__ norm, const float* __restrict__ src,
                       float* __restrict__ dst, float scale) {
  const long total = (long)EE * 32;         // 32 chunks of 4 floats per edge
  long i = (long)blockIdx.x * blockDim.x + threadIdx.x;
  long stride = (long)gridDim.x * blockDim.x;
  for (; i < total; i += stride) {
    int e = (int)(i >> 5);
    int q = (int)(i & 31);
    float s = scale * norm[e];
    long cb = (long)col[e] * BHP + q * 4;
    long rb = (long)row[e] * BHP + q * 4;
    v4f v = *(const v4f*)(src + cb);        // 16B-aligned
#pragma unroll
    for (int u = 0; u < 4; ++u) {
      // skip structural pad lanes (j % 16 == 15): only u==3 when q%4==3
      if (u != 3 || (q & 3) != 3)
        atomicAdd(&dst[rb + u], s * v[u]);
    }
  }
}

// ---------------- WMMA dense accumulate: out[m, g] += Tx[m, 0:16] @ Wp[0:16, g] ----------------
// One wave handles a 16-row x 64-col tile. fp32 WMMA 16x16x4, 4 K-steps, branchless.
__global__ __launch_bounds__(256) void k_gemm_acc(const float* __restrict__ Tx,
                                                  const float* __restrict__ Wp,
                                                  float* __restrict__ outp) {
  int lane = threadIdx.x & 31;
  int wave = threadIdx.x >> 5;
  int mt = blockIdx.x * 8 + wave;       // 16-row tile id in [0, 10000)
  long m0 = (long)mt * 16;
  int lr = lane & 15;
  bool hi = (lane >= 16);

  // A fragments (16x4 fp32): lanes 0-15 hold K={4s,4s+1}, lanes 16-31 hold K={4s+2,4s+3}
  // Padded stride-16 rows -> aligned 8B vector loads.
  const v2f* arow = (const v2f*)Tx + (m0 + lr) * 8 + (hi ? 1 : 0);
  v2f a[4];
#pragma unroll
  for (int s = 0; s < 4; ++s) a[s] = arow[2 * s];

#pragma unroll
  for (int t = 0; t < 4; ++t) {         // four 16-col tiles of G=64
    int col = t * 16 + lr;
    v8f c;
#pragma unroll
    for (int r = 0; r < 8; ++r) {       // C layout: VGPR r -> M=r (lanes 0-15), M=8+r (lanes 16-31)
      long m = m0 + (hi ? 8 + r : r);
      c[r] = outp[m * GG + col];
    }
#pragma unroll
    for (int s = 0; s < 4; ++s) {
      int h0 = 4 * s + (hi ? 2 : 0);
      v2f b;
      b.x = Wp[h0 * GG + col];
      b.y = Wp[(h0 + 1) * GG + col];
      c = __builtin_amdgcn_wmma_f32_16x16x4_f32(false, a[s], false, b,
                                                (short)0, c, false, false);
    }
#pragma unroll
    for (int r = 0; r < 8; ++r) {
      long m = m0 + (hi ? 8 + r : r);
      outp[m * GG + col] = c[r];
    }
  }
}

// ---------------- FC head ----------------

__global__ void k_logits_init(const float* __restrict__ fcb, float* __restrict__ logits) {
  int i = threadIdx.x;
  if (i < BB * CC) logits[i] = fcb[i % CC];
}

// logits[b,c] += sum_{n,g} relu(out[n,b,g] + bias[g]) * fc_w[c, n*G+g]
__global__ __launch_bounds__(256) void k_fc_reduce(const float* __restrict__ outp,
                                                   const float* __restrict__ bias,
                                                   const float* __restrict__ fcw,
                                                   float* __restrict__ logits) {
  float acc[BB * CC];
#pragma unroll
  for (int i = 0; i < BB * CC; ++i) acc[i] = 0.0f;

  long idx = (long)blockIdx.x * blockDim.x + threadIdx.x;
  long stride = (long)gridDim.x * blockDim.x;
  for (; idx < NG; idx += stride) {
    long n = idx >> 6;
    int g = (int)(idx & 63);
    float bg = bias[g];
    float h[BB];
#pragma unroll
    for (int b = 0; b < BB; ++b) {
      float v = outp[n * (BB * GG) + b * GG + g] + bg;
      h[b] = v > 0.0f ? v : 0.0f;
    }
#pragma unroll
    for (int c = 0; c < CC; ++c) {
      float w = fcw[(long)c * NG + idx];
#pragma unroll
      for (int b = 0; b < BB; ++b) acc[b * CC + c] += h[b] * w;
    }
  }

  // wave32 butterfly reduce
#pragma unroll
  for (int i = 0; i < BB * CC; ++i) {
    for (int off = 16; off > 0; off >>= 1)
      acc[i] += __shfl_xor(acc[i], off, 32);
  }

  __shared__ float lds[8 * BB * CC];
  int lane = threadIdx.x & 31, wave = threadIdx.x >> 5;
  if (lane == 0) {
    for (int i = 0; i < BB * CC; ++i) lds[wave * (BB * CC) + i] = acc[i];
  }
  __syncthreads();
  if (threadIdx.x < BB * CC) {
    float s = 0.0f;
    for (int w = 0; w < 8; ++w) s += lds[w * (BB * CC) + threadIdx.x];
    atomicAdd(&logits[threadIdx.x], s);
  }
}

__global__ void k_logsoftmax(const float* __restrict__ logits, float* __restrict__ out) {
  int b = threadIdx.x;
  if (b < BB) {
    float l[CC];
    float m = -1e30f;
    for (int c = 0; c < CC; ++c) { l[c] = logits[b * CC + c]; m = fmaxf(m, l[c]); }
    float s = 0.0f;
    for (int c = 0; c < CC; ++c) s += __expf(l[c] - m);
    float lse = m + __logf(s);
    for (int c = 0; c < CC; ++c) out[b * CC + c] = l[c] - lse;
  }
}

// ---------------- host orchestration ----------------

extern "C" void kernel_launch(void* const* d_in, const int* in_sizes, int n_in,
                              void* d_out, int out_size, void* d_ws, size_t ws_size,
                              hipStream_t stream) {
  const float* x    = (const float*)d_in[0];   // [B,N,H]
  const int*   ei   = (const int*)d_in[1];     // [2,E]
  const float* W    = (const float*)d_in[2];   // [K,H,G]
  const float* bias = (const float*)d_in[3];   // [G]
  const float* fcw  = (const float*)d_in[4];   // [C, N*G]
  const float* fcb  = (const float*)d_in[5];   // [C]
  const int* row = ei;
  const int* col = ei + EE;

  float* ws     = (float*)d_ws;
  float* deg    = ws;                 // N
  float* dinv   = deg    + NN;        // N
  float* norm   = dinv   + NN;        // E
  float* Wp     = norm   + EE;        // K*16*G = 25,600
  float* T0     = Wp     + KK*HP*GG;  // MP
  float* T1     = T0     + MP;        // MP
  float* T2     = T1     + MP;        // MP
  float* outp   = T2     + MP;        // NBG
  float* logits = outp   + NBG;       // B*C
  float* bufs[3] = {T0, T1, T2};

  // 1) degree / normalization / padded weights
  k_zero<<<64, 256, 0, stream>>>(deg, NN);
  k_deg<<<1024, 256, 0, stream>>>(row, deg, EE);
  k_dinv<<<(NN + 255) / 256, 256, 0, stream>>>(deg, dinv, NN);
  k_norm<<<1024, 256, 0, stream>>>(row, col, dinv, norm, EE);
  k_padW<<<(KK * HP * GG + 255) / 256, 256, 0, stream>>>(W, Wp);

  // 2) T0 = transpose(x) -> [N,B,16] padded
  k_transpose<<<4096, 256, 0, stream>>>(x, T0);

  // 3) out = 0; out += T0 @ W[0]
  k_zero<<<8192, 256, 0, stream>>>(outp, NBG);
  k_gemm_acc<<<MM / 16 / 8, 256, 0, stream>>>(T0, Wp, outp);

  // 4) T1 = prop(T0); out += T1 @ W[1]
  k_zero<<<4096, 256, 0, stream>>>(T1, MP);
  k_prop<<<8192, 256, 0, stream>>>(row, col, norm, T0, T1, 1.0f);
  k_gemm_acc<<<MM / 16 / 8, 256, 0, stream>>>(T1, Wp + (long)HP * GG, outp);

  // 5) Chebyshev recursion: Tx_new = 2*prop(Tx) - Tx_prev; out += Tx_new @ W[k]
  for (int k = 2; k < KK; ++k) {
    float* prev = bufs[(k - 2) % 3];
    float* cur  = bufs[(k - 1) % 3];
    float* nxt  = bufs[k % 3];
    k_scalecopy<<<4096, 256, 0, stream>>>(nxt, prev, -1.0f, MP);
    k_prop<<<8192, 256, 0, stream>>>(row, col, norm, cur, nxt, 2.0f);
    k_gemm_acc<<<MM / 16 / 8, 256, 0, stream>>>(nxt, Wp + (long)k * HP * GG, outp);
  }

  // 6) FC head + log_softmax
  k_logits_init<<<1, 64, 0, stream>>>(fcb, logits);
  k_fc_reduce<<<512, 256, 0, stream>>>(outp, bias, fcw, logits);
  k_logsoftmax<<<1, 32, 0, stream>>>(logits, (float*)d_out);
}